// EpisodicMemory_68650757260005
// MI455X (gfx1250) — compile-verified
//
#include <hip/hip_runtime.h>
#include <math.h>

typedef float v2f __attribute__((ext_vector_type(2)));
typedef float v4f __attribute__((ext_vector_type(4)));
typedef float v8f __attribute__((ext_vector_type(8)));

#define NROWS       500000
#define DIM         512
#define VDIM        128
#define NRET        16
#define NTILES      (NROWS / 16)          // 31250
#define WAVES_PER_B 32
#define BLK2        (WAVES_PER_B * 32)    // 1024 threads
#define ROWS_PER_B  (WAVES_PER_B * 16)    // 512 rows
#define NBLK2       ((NTILES + WAVES_PER_B - 1) / WAVES_PER_B)  // 977
#define NCAND       (NBLK2 * NRET)        // 15632
#define NEG_INF     (-3.402823466e38f)

// ---------------------------------------------------------------------------
// Kernel 1: q -> silu(qW1^T+b1)W2^T+b2 -> LayerNorm -> L2 normalize -> qn[512]
// One block of 512 threads; ~2MB of weight traffic, negligible.
// ---------------------------------------------------------------------------
__global__ __launch_bounds__(512) void qproj_kernel(
    const float* __restrict__ query, const float* __restrict__ W1,
    const float* __restrict__ b1, const float* __restrict__ W2,
    const float* __restrict__ b2, const float* __restrict__ gamma,
    const float* __restrict__ beta, float* __restrict__ qn) {
  __shared__ float qv[DIM];
  __shared__ float hv[DIM];
  __shared__ float red[DIM];
  const int t = threadIdx.x;

  qv[t] = query[t];
  __syncthreads();

  // h1 = silu(W1 row_t . q + b1)
  const float* w1r = W1 + (size_t)t * DIM;
  float acc = 0.f;
#pragma clang loop unroll_count(8)
  for (int k = 0; k < DIM; ++k) acc = fmaf(w1r[k], qv[k], acc);
  acc += b1[t];
  float h1 = acc / (1.f + __expf(-acc));   // silu
  hv[t] = h1;
  __syncthreads();

  // h2 = W2 row_t . h1 + b2
  const float* w2r = W2 + (size_t)t * DIM;
  float acc2 = 0.f;
#pragma clang loop unroll_count(8)
  for (int k = 0; k < DIM; ++k) acc2 = fmaf(w2r[k], hv[k], acc2);
  acc2 += b2[t];

  // LayerNorm (biased var) -- block reductions
  red[t] = acc2;
  __syncthreads();
  for (int s = 256; s > 0; s >>= 1) {
    if (t < s) red[t] += red[t + s];
    __syncthreads();
  }
  const float mu = red[0] * (1.f / DIM);
  __syncthreads();

  red[t] = acc2 * acc2;
  __syncthreads();
  for (int s = 256; s > 0; s >>= 1) {
    if (t < s) red[t] += red[t + s];
    __syncthreads();
  }
  const float var = red[0] * (1.f / DIM) - mu * mu;
  __syncthreads();

  const float ln = (acc2 - mu) * rsqrtf(var + 1e-5f) * gamma[t] + beta[t];

  // L2 normalize
  red[t] = ln * ln;
  __syncthreads();
  for (int s = 256; s > 0; s >>= 1) {
    if (t < s) red[t] += red[t + s];
    __syncthreads();
  }
  const float inv = 1.f / fmaxf(sqrtf(red[0]), 1e-12f);
  qn[t] = ln * inv;
}

// ---------------------------------------------------------------------------
// Kernel 2: streaming cosine similarity via V_WMMA_F32_16X16X4_F32
// (dots on matrix pipe, row norms on VALU), then per-block top-16 selection.
// Each wave handles a 16-row tile; 32 waves -> 512 rows per block.
// ---------------------------------------------------------------------------
__global__ __launch_bounds__(BLK2) void sim_topk_kernel(
    const float* __restrict__ keys, const float* __restrict__ qn,
    float* __restrict__ cand_v, int* __restrict__ cand_i) {
  __shared__ float qsm[DIM];
  __shared__ float sims[ROWS_PER_B];
  __shared__ float rv[ROWS_PER_B];
  __shared__ int   rp[ROWS_PER_B];

  const int t = threadIdx.x;
  if (t < DIM) qsm[t] = qn[t];
  __syncthreads();

  const int lane = t & 31;
  const int wave = t >> 5;
  const int tile = blockIdx.x * WAVES_PER_B + wave;
  const bool active = (tile < NTILES);
  const int tsafe = active ? tile : 0;   // keep EXEC all-1s for WMMA

  const int r = lane & 15;   // row within tile (A-matrix M)
  const int h = lane >> 4;   // K-pair half (A-matrix K group)

  const float* rowp = keys + (size_t)(tsafe * 16 + r) * DIM + 4 * h;
  const float* qp   = qsm + 4 * h;

  float nacc = 0.f;   // per-lane partial sum of k^2 for row r
  v8f c = {};         // 16x16 f32 accumulator (dot products, all cols equal)

#pragma clang loop unroll_count(4)
  for (int kk = 0; kk < DIM; kk += 8) {
    // 1 GB total stream, no reuse (>> 192MB L2): non-temporal b128 loads
    v4f a4 = __builtin_nontemporal_load((const v4f*)(rowp + kk));
    v4f b4 = *(const v4f*)(qp + kk);

    v2f a_lo = {a4.x, a4.y};
    v2f b_lo = {b4.x, b4.y};
    v2f a_hi = {a4.z, a4.w};
    v2f b_hi = {b4.z, b4.w};

    // D = A x B + C : A = 16 key rows x 4 cols, B = qn chunk broadcast to
    // all 16 columns (per-lane K mapping identical to A -> consistent sum).
    c = __builtin_amdgcn_wmma_f32_16x16x4_f32(
        false, a_lo, false, b_lo, (short)0, c, false, false);
    c = __builtin_amdgcn_wmma_f32_16x16x4_f32(
        false, a_hi, false, b_hi, (short)0, c, false, false);

    // row-norm accumulation co-executes on the VALU
    nacc = fmaf(a4.x, a4.x, nacc);
    nacc = fmaf(a4.y, a4.y, nacc);
    nacc = fmaf(a4.z, a4.z, nacc);
    nacc = fmaf(a4.w, a4.w, nacc);
  }

  // combine the two K-halves of each row's norm: lane l <-> lane l^16
  nacc += __shfl_xor(nacc, 16, 32);
  // now lane l holds ||key_row(l&15)||^2 in full

  // D layout: lane (h half) VGPR m holds row 8h+m, col = r. Use col 0 lanes.
#pragma unroll
  for (int m = 0; m < 8; ++m) {
    const int rowm = 8 * h + m;
    const float n2 = __shfl(nacc, rowm, 32);          // norm of row 8h+m
    const float s = c[m] * rsqrtf(fmaxf(n2, 1e-24f)); // qn is unit-length
    if (r == 0) sims[wave * 16 + rowm] = active ? s : NEG_INF;
  }
  __syncthreads();

  // block top-16 over 512 sims: 16 rounds of argmax-with-exclusion
  for (int it = 0; it < NRET; ++it) {
    if (t < ROWS_PER_B) { rv[t] = sims[t]; rp[t] = t; }
    __syncthreads();
    for (int s = ROWS_PER_B / 2; s > 0; s >>= 1) {
      if (t < s) {
        if (rv[t + s] > rv[t]) { rv[t] = rv[t + s]; rp[t] = rp[t + s]; }
      }
      __syncthreads();
    }
    if (t == 0) {
      cand_v[blockIdx.x * NRET + it] = rv[0];
      cand_i[blockIdx.x * NRET + it] = blockIdx.x * ROWS_PER_B + rp[0];
      sims[rp[0]] = NEG_INF;
    }
    __syncthreads();
  }
}

// ---------------------------------------------------------------------------
// Kernel 3: reduce 15632 candidates -> global top-16, softmax, gather values,
// weighted sum -> out[128]. Single block; candidates are L2-resident.
// ---------------------------------------------------------------------------
__global__ __launch_bounds__(1024) void finalize_kernel(
    float* __restrict__ cand_v, const int* __restrict__ cand_i, int ncand,
    const float* __restrict__ values, float* __restrict__ out) {
  __shared__ float rv[1024];
  __shared__ int   rp[1024];
  __shared__ float topv[NRET];
  __shared__ int   topi[NRET];
  __shared__ float attn[NRET];
  const int t = threadIdx.x;

  for (int it = 0; it < NRET; ++it) {
    float best = NEG_INF;
    int bp = 0;
    for (int i = t; i < ncand; i += 1024) {
      const float v = cand_v[i];
      if (v > best) { best = v; bp = i; }
    }
    rv[t] = best; rp[t] = bp;
    __syncthreads();
    for (int s = 512; s > 0; s >>= 1) {
      if (t < s) {
        if (rv[t + s] > rv[t]) { rv[t] = rv[t + s]; rp[t] = rp[t + s]; }
      }
      __syncthreads();
    }
    if (t == 0) {
      topv[it] = rv[0];
      topi[it] = cand_i[rp[0]];
      cand_v[rp[0]] = NEG_INF;   // exclude; cand_v rewritten by kernel2 each call
    }
    __syncthreads();
  }

  if (t == 0) {
    const float m = topv[0];   // sorted descending -> max first
    float ssum = 0.f;
    for (int k = 0; k < NRET; ++k) {
      const float e = __expf(topv[k] - m);
      attn[k] = e;
      ssum += e;
    }
    const float inv = 1.f / ssum;
    for (int k = 0; k < NRET; ++k) attn[k] *= inv;
  }
  __syncthreads();

  if (t < VDIM) {
    float acc = 0.f;
#pragma unroll
    for (int k = 0; k < NRET; ++k)
      acc = fmaf(attn[k], values[(size_t)topi[k] * VDIM + t], acc);
    out[t] = acc;
  }
}

// ---------------------------------------------------------------------------
extern "C" void kernel_launch(void* const* d_in, const int* in_sizes, int n_in,
                              void* d_out, int out_size, void* d_ws,
                              size_t ws_size, hipStream_t stream) {
  const float* query  = (const float*)d_in[0];
  const float* W1     = (const float*)d_in[1];
  const float* b1     = (const float*)d_in[2];
  const float* W2     = (const float*)d_in[3];
  const float* b2     = (const float*)d_in[4];
  const float* gamma  = (const float*)d_in[5];
  const float* beta   = (const float*)d_in[6];
  const float* keys   = (const float*)d_in[7];
  const float* values = (const float*)d_in[8];

  float* ws     = (float*)d_ws;
  float* qn     = ws;                    // 512 floats
  float* cand_v = ws + DIM;              // NCAND floats
  int*   cand_i = (int*)(ws + DIM + NCAND);

  qproj_kernel<<<1, DIM, 0, stream>>>(query, W1, b1, W2, b2, gamma, beta, qn);
  sim_topk_kernel<<<NBLK2, BLK2, 0, stream>>>(keys, qn, cand_v, cand_i);
  finalize_kernel<<<1, 1024, 0, stream>>>(cand_v, cand_i, NCAND, values,
                                          (float*)d_out);
}